// CrossNet_74569222193586
// MI455X (gfx1250) — compile-verified
//
#include <hip/hip_runtime.h>

// CrossNet fused kernel for MI455X (gfx1250), fp32, HBM-bandwidth-bound.
// out = X * (X @ alphas + 1) + bias

typedef __attribute__((ext_vector_type(2))) float v2f;
typedef __attribute__((ext_vector_type(4))) float v4f;
typedef __attribute__((ext_vector_type(8))) float v8f;

#define DDIM    4096
#define ROWS    16               // rows per block (one WMMA M-tile)
#define CHUNK   256              // K columns staged per chunk
#define NCHUNK  (DDIM / CHUNK)   // 16
#define PAD     4                // LDS row padding -> conflict-free b64 reads
#define XSTR    (CHUNK + PAD)    // 260 floats
#define THREADS 256              // 8 waves (wave32)

__global__ __launch_bounds__(THREADS)
void crossnet_wmma_kernel(const float* __restrict__ X,
                          const float* __restrict__ alphas,
                          const float* __restrict__ bias,
                          float* __restrict__ out,
                          int nrows)
{
    __shared__ __align__(16) float sX[2][ROWS * XSTR]; // 33,280 B
    __shared__ __align__(16) float sAlpha[DDIM];       // 16,384 B
    __shared__ float sPart[8 * ROWS];                  // per-wave row partials
    __shared__ float sScale[ROWS];

    const int t    = threadIdx.x;
    const int wave = t >> 5;
    const int lane = t & 31;
    const size_t rowBase = (size_t)blockIdx.x * ROWS;

    const v4f* X4 = reinterpret_cast<const v4f*>(X);
    const v4f* A4 = reinterpret_cast<const v4f*>(alphas);
    const v4f* B4 = reinterpret_cast<const v4f*>(bias);
    v4f*       O4 = reinterpret_cast<v4f*>(out);

    // ---- stage alphas (16 KB) once, coalesced b128 ----
    for (int i = t; i < DDIM / 4; i += THREADS)
        *reinterpret_cast<v4f*>(&sAlpha[i * 4]) = A4[i];

    // ---- stage chunk 0 of the 16-row X tile ----
    {
#pragma unroll
        for (int j = 0; j < 4; ++j) {
            int s   = t + j * THREADS;      // 0..1023 float4 slots
            int row = s >> 6;               // /64
            int c4  = s & 63;
            v4f v = X4[((rowBase + row) * DDIM + c4 * 4) >> 2];
            *reinterpret_cast<v4f*>(&sX[0][row * XSTR + c4 * 4]) = v;
        }
    }
    __syncthreads();

    // WMMA fragment indexing (ISA 32-bit 16x4 A/B layout, wave32):
    //   lanes 0-15 : M = lane, VGPR0=K0, VGPR1=K1
    //   lanes 16-31: M = lane-16, VGPR0=K2, VGPR1=K3
    const int m    = lane & 15;
    const int koff = (lane < 16) ? 0 : 2;
    const int kw   = wave * 32;             // this wave's K strip in chunk

    v8f acc = {};                            // D[m][n] = partial dot(row m)

    for (int c = 0; c < NCHUNK; ++c) {
        const int cur = c & 1;

        // issue next chunk's global loads early (overlap with WMMA below)
        v4f r[4];
        if (c + 1 < NCHUNK) {
#pragma unroll
            for (int j = 0; j < 4; ++j) {
                int s   = t + j * THREADS;
                int row = s >> 6;
                int c4  = s & 63;
                r[j] = X4[((rowBase + row) * DDIM + (size_t)(c + 1) * CHUNK + c4 * 4) >> 2];
            }
        }

        // exact-f32 GEMV reduction: 8x V_WMMA_F32_16X16X4_F32 per wave/chunk
        const float* xb = sX[cur];
#pragma unroll
        for (int k = 0; k < 32; k += 4) {
            v2f a = *reinterpret_cast<const v2f*>(&xb[m * XSTR + kw + k + koff]);
            v2f b = *reinterpret_cast<const v2f*>(&sAlpha[c * CHUNK + kw + k + koff]);
            acc = __builtin_amdgcn_wmma_f32_16x16x4_f32(
                false, a, false, b, (short)0, acc, false, false);
        }

        // commit staged data to the other buffer, then barrier
        if (c + 1 < NCHUNK) {
            float* xn = sX[cur ^ 1];
#pragma unroll
            for (int j = 0; j < 4; ++j) {
                int s   = t + j * THREADS;
                int row = s >> 6;
                int c4  = s & 63;
                *reinterpret_cast<v4f*>(&xn[row * XSTR + c4 * 4]) = r[j];
            }
        }
        __syncthreads();
    }

    // ---- combine 8 waves' partials: D layout: VGPR p, lane0 -> row p, lane16 -> row p+8
    if (lane == 0 || lane == 16) {
        const int half = (lane >> 4) * 8;
#pragma unroll
        for (int p = 0; p < 8; ++p)
            sPart[wave * ROWS + half + p] = acc[p];
    }
    __syncthreads();

    if (t < ROWS) {
        float s = 0.0f;
#pragma unroll
        for (int w = 0; w < 8; ++w) s += sPart[w * ROWS + t];
        sScale[t] = s + 1.0f;                // xa + 1 (folds the "+ X" term)
    }
    __syncthreads();

    // ---- output pass: out = X * scale + bias  (X re-read hits L2) ----
    for (int i = t; i < ROWS * (DDIM / 4); i += THREADS) {
        int row = i >> 10;                   // / (DDIM/4)
        int c4  = i & 1023;
        size_t g = ((rowBase + row) * DDIM >> 2) + c4;
        v4f x4 = X4[g];
        v4f b4 = B4[c4];
        float sc = sScale[row];
        O4[g] = x4 * sc + b4;
    }
}

extern "C" void kernel_launch(void* const* d_in, const int* in_sizes, int n_in,
                              void* d_out, int out_size, void* d_ws, size_t ws_size,
                              hipStream_t stream) {
    const float* X      = (const float*)d_in[0];
    const float* alphas = (const float*)d_in[1];
    const float* bias   = (const float*)d_in[2];
    float* out          = (float*)d_out;

    const int B = in_sizes[0] / DDIM;        // 16384
    dim3 grid(B / ROWS);                     // 1024 blocks of 16 rows
    crossnet_wmma_kernel<<<grid, THREADS, 0, stream>>>(X, alphas, bias, out, B);
}